// AttentiveMeanPooler_53687091200465
// MI455X (gfx1250) — compile-verified
//
#include <hip/hip_runtime.h>
#include <hip/hip_bf16.h>

#define BATCH 16
#define SDIM  4096
#define HDIM  256
#define DDIM  255   // H-1 output columns of W

typedef __attribute__((ext_vector_type(2))) float v2f;
typedef __attribute__((ext_vector_type(8))) float v8f;

// ---------------------------------------------------------------------------
// Kernel 0: repack Wkv (256 x 255, row-major) into zero-padded, B-fragment-
// ordered Wf so each lane's 16x16x4 B fragment is one aligned b64 load:
//   k = 4g + 2h + p  ->  Wf[((g*2 + h)*256 + j)*2 + p]
// Lane reads pair (W[4g+2h][j], W[4g+2h+1][j]) at  (g*2 + h)*512 + j*2.
// ---------------------------------------------------------------------------
__global__ void prep_w_kernel(const float* __restrict__ Wkv,
                              float* __restrict__ Wf)
{
    const int k = blockIdx.x;      // 0..255
    const int j = threadIdx.x;     // 0..255
    const float v = (j < DDIM) ? Wkv[k * DDIM + j] : 0.0f;
    const int g = k >> 2;
    const int h = (k >> 1) & 1;
    const int p = k & 1;
    Wf[((g * 2 + h) * 256 + j) * 2 + p] = v;
}

// ---------------------------------------------------------------------------
// Kernel 1: query = lorentz_linear(cls, Wq, bq), stored with signs applied:
// qsigned[b] = [-t, y_0 .. y_254]
// ---------------------------------------------------------------------------
__global__ void query_kernel(const float* __restrict__ hidden,
                             const float* __restrict__ Wq,
                             const float* __restrict__ bq,
                             float* __restrict__ qsigned)
{
    __shared__ float clsS[HDIM];
    __shared__ float red[HDIM];
    const int b = blockIdx.x;
    const int tid = threadIdx.x;

    clsS[tid] = hidden[(size_t)b * SDIM * HDIM + tid];
    __syncthreads();

    float y = 0.0f;
    if (tid < DDIM) {
        float acc = bq[tid];
        #pragma unroll 4
        for (int k = 0; k < HDIM; ++k)
            acc = fmaf(clsS[k], Wq[k * DDIM + tid], acc);
        y = acc;
    }
    red[tid] = y * y;   // tid==255 contributes 0
    __syncthreads();
    for (int off = 128; off > 0; off >>= 1) {
        if (tid < off) red[tid] += red[tid + off];
        __syncthreads();
    }
    if (tid == 0)    qsigned[b * HDIM]           = -sqrtf(red[0] + 1.0f);
    if (tid < DDIM)  qsigned[b * HDIM + 1 + tid] = y;
}

// ---------------------------------------------------------------------------
// Kernel 2: kv = lorentz_linear(hidden, Wkv, bkv)  (fp32 WMMA GEMM)
//           + logits[b][s] = qsigned[b] . kv[b][s]
// One block per 16-row tile; 8 waves x two 16-col tiles each; K in steps of 4
// with V_WMMA_F32_16X16X4_F32. B fragments come from the padded Wf layout
// (one b64 per tile per step, branch-free); K loop software-pipelined by 1.
// ---------------------------------------------------------------------------
__global__ void kv_logits_kernel(const float* __restrict__ hidden,
                                 const float* __restrict__ Wf,
                                 const float* __restrict__ bkv,
                                 const float* __restrict__ qsigned,
                                 float* __restrict__ kv,
                                 float* __restrict__ logits)
{
    __shared__ float Xs[16][260];   // input tile, padded stride (bank-safe)
    __shared__ float Ys[16][260];   // output tile: col0 = t, cols 1..255 = y
    __shared__ float qsh[HDIM];
    __shared__ float rowacc[16];

    const int tid  = threadIdx.x;
    const int lane = tid & 31;
    const int wave = tid >> 5;
    const int r0 = blockIdx.x << 4;          // global row base (0..65520)
    const int b  = r0 >> 12;                 // batch index
    const int s0 = r0 & (SDIM - 1);          // sequence offset within batch

    qsh[tid] = qsigned[b * HDIM + tid];
    if (tid < 16) rowacc[tid] = 0.0f;

    // --- load 16x256 X tile, coalesced float4 ---
    const float* src = hidden + (size_t)r0 * HDIM;
    #pragma unroll
    for (int it = 0; it < 4; ++it) {
        int f4  = tid + it * 256;            // 0..1023
        int e   = f4 << 2;
        int row = e >> 8;
        int col = e & 255;
        const float4 v = reinterpret_cast<const float4*>(src)[f4];
        Xs[row][col + 0] = v.x;
        Xs[row][col + 1] = v.y;
        Xs[row][col + 2] = v.z;
        Xs[row][col + 3] = v.w;
    }
    __syncthreads();

    // Fragment index math per ISA layouts:
    //  A 16x4 f32 : lanes 0-15 -> M=lane, K=kk+{0,1}; lanes 16-31 -> K=kk+{2,3}
    //  B 4x16 f32 : VGPR g, lanes 0-15 -> K=kk+g, N=lane; lanes 16-31 -> K=kk+g+2
    //  C 16x16 f32: VGPR g, lanes 0-15 -> M=g; lanes 16-31 -> M=g+8
    const int mA    = lane & 15;
    const int hi    = lane >> 4;             // 0 or 1 (which K half-pair)
    const int khalf = hi << 1;               // 0 or 2
    const int j0a = wave << 5;               // this wave's first col tile
    const int ja  = j0a + mA;                // column of tile a  (<= 239)
    const int jb  = j0a + 16 + mA;           // column of tile b  (<= 255, padded)

    const float* xrow = &Xs[mA][khalf];              // += 4 per K group
    const float* wpa  = Wf + hi * 512 + ja * 2;      // += 1024 per K group
    const float* wpb  = Wf + hi * 512 + jb * 2;

    v8f ca = {0.f,0.f,0.f,0.f,0.f,0.f,0.f,0.f};
    v8f cb = {0.f,0.f,0.f,0.f,0.f,0.f,0.f,0.f};

    // software-pipelined K loop (64 groups of K=4)
    v2f a0  = *(const v2f*)(xrow);
    v2f wa0 = *(const v2f*)(wpa);
    v2f wb0 = *(const v2f*)(wpb);
    for (int g = 0; g < 63; ++g) {
        const v2f a1  = *(const v2f*)(xrow + (g + 1) * 4);
        const v2f wa1 = *(const v2f*)(wpa + (g + 1) * 1024);
        const v2f wb1 = *(const v2f*)(wpb + (g + 1) * 1024);
        ca = __builtin_amdgcn_wmma_f32_16x16x4_f32(false, a0, false, wa0,
                                                   (short)0, ca, false, false);
        cb = __builtin_amdgcn_wmma_f32_16x16x4_f32(false, a0, false, wb0,
                                                   (short)0, cb, false, false);
        a0 = a1; wa0 = wa1; wb0 = wb1;
    }
    ca = __builtin_amdgcn_wmma_f32_16x16x4_f32(false, a0, false, wa0,
                                               (short)0, ca, false, false);
    cb = __builtin_amdgcn_wmma_f32_16x16x4_f32(false, a0, false, wb0,
                                               (short)0, cb, false, false);

    // --- bias, stash y into Ys, accumulate per-row sum of squares ---
    // (pad column jb==255: weights & bias are 0 => yb==0; Ys[m][256] is in-range
    //  scratch and never read back, so everything below is branch-free)
    const int mBase = hi << 3;               // 0 or 8
    const float biasa = bkv[ja];
    const float biasb = (jb < DDIM) ? bkv[jb] : 0.0f;
    #pragma unroll
    for (int g = 0; g < 8; ++g) {
        const int m = mBase + g;
        const float ya = ca[g] + biasa;
        const float yb = cb[g] + biasb;
        Ys[m][1 + ja] = ya;
        Ys[m][1 + jb] = yb;
        atomicAdd(&rowacc[m], ya * ya + yb * yb);   // ds_add_f32
    }
    __syncthreads();

    if (tid < 16) {
        Ys[tid][0] = sqrtf(rowacc[tid] + 1.0f);   // time component t
        rowacc[tid] = 0.0f;                       // reuse for logits
    }
    __syncthreads();

    // --- coalesced kv tile store (16 rows x 256) ---
    float* dst = kv + (size_t)r0 * HDIM;
    #pragma unroll
    for (int m = 0; m < 16; ++m)
        dst[m * HDIM + tid] = Ys[m][tid];

    // --- logits: 16 threads per row, 16 elems each ---
    {
        const int m = tid >> 4;
        const int chunk = (tid & 15) << 4;
        float p = 0.0f;
        #pragma unroll
        for (int i = 0; i < 16; ++i)
            p = fmaf(qsh[chunk + i], Ys[m][chunk + i], p);
        atomicAdd(&rowacc[m], p);
    }
    __syncthreads();
    if (tid < 16)
        logits[b * SDIM + s0 + tid] = rowacc[tid];
}

// ---------------------------------------------------------------------------
// Kernel 3: masked softmax over k (one block per batch)
// ---------------------------------------------------------------------------
__global__ void softmax_kernel(const float* __restrict__ logits,
                               const unsigned char* __restrict__ mask,
                               float* __restrict__ attn)
{
    __shared__ float red[256];
    const int b = blockIdx.x;
    const int tid = threadIdx.x;
    const float* lg = logits + b * SDIM;
    const unsigned char* mk = mask + b * SDIM;

    float mx = -INFINITY;
    for (int k = tid; k < SDIM; k += 256)
        mx = fmaxf(mx, mk[k] ? lg[k] : -INFINITY);
    red[tid] = mx;
    __syncthreads();
    for (int off = 128; off > 0; off >>= 1) {
        if (tid < off) red[tid] = fmaxf(red[tid], red[tid + off]);
        __syncthreads();
    }
    mx = red[0];
    __syncthreads();

    float sum = 0.0f;
    for (int k = tid; k < SDIM; k += 256)
        sum += mk[k] ? expf(lg[k] - mx) : 0.0f;
    red[tid] = sum;
    __syncthreads();
    for (int off = 128; off > 0; off >>= 1) {
        if (tid < off) red[tid] += red[tid + off];
        __syncthreads();
    }
    const float inv = 1.0f / red[0];

    for (int k = tid; k < SDIM; k += 256)
        attn[b * SDIM + k] = mk[k] ? expf(lg[k] - mx) * inv : 0.0f;
}

// ---------------------------------------------------------------------------
// Kernel 4: s = attn @ kv, Lorentz-normalize, write out[b][0..255]
// kv is hot in the 192MB L2, reads fully coalesced (thread = column d).
// ---------------------------------------------------------------------------
__global__ void pool_kernel(const float* __restrict__ kv,
                            const float* __restrict__ attn,
                            float* __restrict__ out)
{
    __shared__ float ach[256];
    __shared__ float red[256];
    const int b = blockIdx.x;
    const int tid = threadIdx.x;
    const float* kvb = kv + (size_t)b * SDIM * HDIM;
    const float* ab  = attn + b * SDIM;

    float acc = 0.0f;
    for (int kb = 0; kb < SDIM; kb += 256) {
        __syncthreads();
        ach[tid] = ab[kb + tid];
        __syncthreads();
        #pragma unroll 4
        for (int i = 0; i < 256; ++i)
            acc = fmaf(ach[i], kvb[(size_t)(kb + i) * HDIM + tid], acc);
    }

    // sq_norm = -sum(signs * s^2) = s0^2 - sum_{d>0} s_d^2
    red[tid] = (tid == 0) ? -acc * acc : acc * acc;
    __syncthreads();
    for (int off = 128; off > 0; off >>= 1) {
        if (tid < off) red[tid] += red[tid + off];
        __syncthreads();
    }
    const float sq_norm = -red[0];
    const float denom = sqrtf(fmaxf(sq_norm, 1e-8f));
    out[b * HDIM + tid] = acc / denom;
}

// ---------------------------------------------------------------------------
extern "C" void kernel_launch(void* const* d_in, const int* in_sizes, int n_in,
                              void* d_out, int out_size, void* d_ws, size_t ws_size,
                              hipStream_t stream)
{
    const float*         hidden = (const float*)d_in[0];
    const unsigned char* mask   = (const unsigned char*)d_in[1]; // jnp bool = 1 byte
    const float*         Wq     = (const float*)d_in[2];
    const float*         bq     = (const float*)d_in[3];
    const float*         Wkv    = (const float*)d_in[4];
    const float*         bkv    = (const float*)d_in[5];
    float* out = (float*)d_out;

    // Workspace layout (floats): kv | qsigned | logits | attn | Wf
    float* ws      = (float*)d_ws;
    float* kv      = ws;                                        // 16*4096*256
    float* qsigned = ws + (size_t)BATCH * SDIM * HDIM;          // 16*256
    float* logits  = qsigned + BATCH * HDIM;                    // 16*4096
    float* attn    = logits + BATCH * SDIM;                     // 16*4096
    float* Wf      = attn + BATCH * SDIM;                       // 256*256 padded

    prep_w_kernel  <<<HDIM,              256, 0, stream>>>(Wkv, Wf);
    query_kernel   <<<BATCH,             256, 0, stream>>>(hidden, Wq, bq, qsigned);
    kv_logits_kernel<<<BATCH * SDIM / 16, 256, 0, stream>>>(hidden, Wf, bkv,
                                                            qsigned, kv, logits);
    softmax_kernel <<<BATCH,             256, 0, stream>>>(logits, mask, attn);
    pool_kernel    <<<BATCH,             256, 0, stream>>>(kv, attn, out);
}